// WeightedActor_13469017441101
// MI455X (gfx1250) — compile-verified
//
#include <hip/hip_runtime.h>
#include <math.h>

#define N_SAMPLES 524288
#define N_ACTORS  8
#define DIM_D     64
#define DIM_E     32
#define LOG2PI    1.8378770664093453f

#define SORT_BLOCK   256
#define SORT_NBLOCKS (N_SAMPLES / SORT_BLOCK)   // 2048
#define SCAN_TOTAL   (N_ACTORS * SORT_NBLOCKS)  // 16384

#define TILES        (N_SAMPLES / 16)           // 32768
#define WAVES_PER_BLOCK 8
#define TILE_BLOCKS  (TILES / WAVES_PER_BLOCK)  // 4096

typedef __attribute__((ext_vector_type(2))) float v2f;
typedef __attribute__((ext_vector_type(8))) float v8f;

// ---------------------------------------------------------------------------
// Sort pass 1: per-element stable local rank within block + per-block counts.
// ---------------------------------------------------------------------------
__global__ __launch_bounds__(SORT_BLOCK)
void sort_count_kernel(const int* __restrict__ actor_idx,
                       int* __restrict__ rank_local,
                       int* __restrict__ block_counts)
{
    __shared__ int waveCnt[8][N_ACTORS];   // [wave][actor]
    const int t    = threadIdx.x;
    const int lane = t & 31;
    const int w    = t >> 5;
    const int i    = blockIdx.x * SORT_BLOCK + t;
    const int a    = actor_idx[i];

    unsigned long long lowmask = (1ull << lane) - 1ull;
    int myrank = 0;
    #pragma unroll
    for (int aa = 0; aa < N_ACTORS; ++aa) {
        unsigned long long bits = __ballot(a == aa);
        if (lane == 0) waveCnt[w][aa] = (int)__popcll(bits);
        if (aa == a)   myrank = (int)__popcll(bits & lowmask);
    }
    __syncthreads();

    int before = 0;
    for (int w2 = 0; w2 < w; ++w2) before += waveCnt[w2][a];
    rank_local[i] = before + myrank;

    if (t < N_ACTORS) {   // t == actor id
        int c = 0;
        #pragma unroll
        for (int w2 = 0; w2 < 8; ++w2) c += waveCnt[w2][t];
        block_counts[t * SORT_NBLOCKS + blockIdx.x] = c;
    }
}

// ---------------------------------------------------------------------------
// Sort pass 2: single-block exclusive scan of the flattened [A][NBLOCKS]
// count matrix (actor-major order == stable counting-sort base offsets).
// ---------------------------------------------------------------------------
__global__ __launch_bounds__(256)
void sort_scan_kernel(int* __restrict__ data)
{
    const int PER = SCAN_TOTAL / 256;   // 64
    __shared__ int partial[256];
    const int t = threadIdx.x;
    const int base = t * PER;

    int s = 0;
    for (int f = 0; f < PER; ++f) s += data[base + f];
    partial[t] = s;
    __syncthreads();

    // Hillis-Steele inclusive scan over 256 partials
    for (int off = 1; off < 256; off <<= 1) {
        int v = (t >= off) ? partial[t - off] : 0;
        __syncthreads();
        partial[t] += v;
        __syncthreads();
    }
    int running = partial[t] - s;       // exclusive base for this thread
    for (int f = 0; f < PER; ++f) {
        int tmp = data[base + f];
        data[base + f] = running;
        running += tmp;
    }
}

// ---------------------------------------------------------------------------
// Sort pass 3: scatter -> p[pos] = i  (stable).
// ---------------------------------------------------------------------------
__global__ __launch_bounds__(SORT_BLOCK)
void sort_scatter_kernel(const int* __restrict__ actor_idx,
                         const int* __restrict__ rank_local,
                         const int* __restrict__ block_offsets,
                         int* __restrict__ p)
{
    const int i = blockIdx.x * SORT_BLOCK + threadIdx.x;
    const int a = actor_idx[i];
    const int pos = block_offsets[a * SORT_NBLOCKS + blockIdx.x] + rank_local[i];
    p[pos] = i;
}

// ---------------------------------------------------------------------------
// Main compute: one wave32 per 16-row output tile.
//   out_action[i, :] = state[p[p[i]]] @ W[a] + b[a] + exp(log_std[a]) * eps[p[i]]
// fp32 WMMA: D = A(16x4) x B(4x16) + C, 16 K-steps over D=64, 2 N-tiles (E=32).
// ---------------------------------------------------------------------------
__global__ __launch_bounds__(256)
void actor_tiles_kernel(const float* __restrict__ state,
                        const float* __restrict__ Wg,
                        const float* __restrict__ bg,
                        const float* __restrict__ log_std_g,
                        const float* __restrict__ eps,
                        const int*   __restrict__ actor_idx,
                        const int*   __restrict__ p,
                        float*       __restrict__ out_action)
{
    __shared__ float Wl[N_ACTORS * DIM_D * DIM_E];   // 64 KB
    __shared__ float bl[N_ACTORS * DIM_E];
    __shared__ float lsl[N_ACTORS * DIM_E];

    const int t = threadIdx.x;
    // Stage all actors' W / b / log_std into LDS (float4 cooperative copy).
    for (int f = t; f < (N_ACTORS * DIM_D * DIM_E) / 4; f += 256)
        ((float4*)Wl)[f] = ((const float4*)Wg)[f];
    for (int f = t; f < N_ACTORS * DIM_E; f += 256) {
        bl[f]  = bg[f];
        lsl[f] = log_std_g[f];
    }
    __syncthreads();

    const int wave = t >> 5;
    const int lane = t & 31;
    const int l15  = lane & 15;
    const int h    = lane >> 4;             // lane half: K-split for A/B frags
    const int tile = blockIdx.x * WAVES_PER_BLOCK + wave;

    const int i = tile * 16 + l15;          // output row handled by this lane-slot
    const int k = p[i];                     // sorted position
    const int j = p[k];                     // original sample index
    const int a_row = actor_idx[j];

    // A fragments: 16 K-steps, each v2f per ISA 16x4 fp32 layout
    //   vgpr0 = A[M=l15][4kk + 2h], vgpr1 = A[M=l15][4kk + 2h + 1]
    const float* srow = state + (size_t)j * DIM_D + 2 * h;
    v2f afrag[16];
    #pragma unroll
    for (int kk = 0; kk < 16; ++kk) {
        afrag[kk].x = srow[kk * 4 + 0];
        afrag[kk].y = srow[kk * 4 + 1];
    }

    // Tile actor range (wave-uniform; sorted groups are contiguous).
    int amin = a_row, amax = a_row;
    #pragma unroll
    for (int off = 16; off >= 1; off >>= 1) {
        amin = min(amin, __shfl_xor(amin, off, 32));
        amax = max(amax, __shfl_xor(amax, off, 32));
    }

    v8f m0 = {}; v8f m1 = {};               // selected means, N-tiles 0/1
    for (int a = amin; a <= amax; ++a) {    // uniform loop
        const float* Wa = Wl + a * (DIM_D * DIM_E);
        v8f c0 = {}; v8f c1 = {};
        #pragma unroll
        for (int kk = 0; kk < 16; ++kk) {
            const int krow = kk * 4 + 2 * h;
            v2f b0, b1;
            b0.x = Wa[(krow    ) * DIM_E + l15];
            b0.y = Wa[(krow + 1) * DIM_E + l15];
            b1.x = Wa[(krow    ) * DIM_E + 16 + l15];
            b1.y = Wa[(krow + 1) * DIM_E + 16 + l15];
            c0 = __builtin_amdgcn_wmma_f32_16x16x4_f32(
                     false, afrag[kk], false, b0, (short)0, c0, false, false);
            c1 = __builtin_amdgcn_wmma_f32_16x16x4_f32(
                     false, afrag[kk], false, b1, (short)0, c1, false, false);
        }
        #pragma unroll
        for (int r = 0; r < 8; ++r) {
            const int rowM = r + 8 * h;
            const int a_of_row = __shfl(a_row, rowM, 32);
            if (a_of_row == a) { m0[r] = c0[r]; m1[r] = c1[r]; }
        }
    }

    // Epilogue: + b[a] + exp(log_std[a]) * eps[k], store row-by-row.
    #pragma unroll
    for (int r = 0; r < 8; ++r) {
        const int rowM = r + 8 * h;
        const int k_r  = __shfl(k, rowM, 32);
        const int a_r  = __shfl(a_row, rowM, 32);
        const int i_r  = tile * 16 + rowM;

        const float e0  = eps[(size_t)k_r * DIM_E + l15];
        const float e1  = eps[(size_t)k_r * DIM_E + 16 + l15];
        const float st0 = __expf(lsl[a_r * DIM_E + l15]);
        const float st1 = __expf(lsl[a_r * DIM_E + 16 + l15]);
        out_action[(size_t)i_r * DIM_E + l15]      = m0[r] + bl[a_r * DIM_E + l15]      + st0 * e0;
        out_action[(size_t)i_r * DIM_E + 16 + l15] = m1[r] + bl[a_r * DIM_E + 16 + l15] + st1 * e1;
    }
}

// ---------------------------------------------------------------------------
// log-prob: one wave per output row; (action-mean)/std == eps exactly, so
//   lp[i] = sum_e(-0.5*eps[k,e]^2 - log_std[a,e]) - 16*log(2*pi)
// ---------------------------------------------------------------------------
__global__ __launch_bounds__(256)
void lp_kernel(const float* __restrict__ eps,
               const float* __restrict__ log_std_g,
               const int*   __restrict__ actor_idx,
               const int*   __restrict__ p,
               float*       __restrict__ out_lp)
{
    const int t = threadIdx.x;
    const int wave = t >> 5;
    const int lane = t & 31;
    const int i = blockIdx.x * 8 + wave;

    const int k = p[i];
    const int j = p[k];
    const int a = actor_idx[j];

    const float e  = eps[(size_t)k * DIM_E + lane];
    const float ls = log_std_g[a * DIM_E + lane];
    float v = -0.5f * e * e - ls;
    #pragma unroll
    for (int off = 16; off >= 1; off >>= 1)
        v += __shfl_xor(v, off, 32);
    if (lane == 0)
        out_lp[i] = v - 16.0f * LOG2PI;
}

// ---------------------------------------------------------------------------
extern "C" void kernel_launch(void* const* d_in, const int* in_sizes, int n_in,
                              void* d_out, int out_size, void* d_ws, size_t ws_size,
                              hipStream_t stream)
{
    const float* state     = (const float*)d_in[0];
    const float* W         = (const float*)d_in[1];
    const float* b         = (const float*)d_in[2];
    const float* log_std   = (const float*)d_in[3];
    const float* eps       = (const float*)d_in[4];
    const int*   actor_idx = (const int*)d_in[5];

    float* out_action = (float*)d_out;                       // [N, E]
    float* out_lp     = (float*)d_out + (size_t)N_SAMPLES * DIM_E;  // [N]

    int* rank_local   = (int*)d_ws;                          // N ints
    int* block_counts = rank_local + N_SAMPLES;              // A*NBLOCKS ints
    int* p            = block_counts + SCAN_TOTAL;           // N ints

    sort_count_kernel<<<SORT_NBLOCKS, SORT_BLOCK, 0, stream>>>(
        actor_idx, rank_local, block_counts);
    sort_scan_kernel<<<1, 256, 0, stream>>>(block_counts);
    sort_scatter_kernel<<<SORT_NBLOCKS, SORT_BLOCK, 0, stream>>>(
        actor_idx, rank_local, block_counts, p);

    actor_tiles_kernel<<<TILE_BLOCKS, 256, 0, stream>>>(
        state, W, b, log_std, eps, actor_idx, p, out_action);

    lp_kernel<<<N_SAMPLES / 8, 256, 0, stream>>>(
        eps, log_std, actor_idx, p, out_lp);
}